// LeagueGNN_14207751815591
// MI455X (gfx1250) — compile-verified
//
#include <hip/hip_runtime.h>
#include <math.h>

typedef __attribute__((ext_vector_type(2))) float v2f;
typedef __attribute__((ext_vector_type(8))) float v8f;

#define NNODES 500000
#define NEDGES 4000000
#define NGRAPH 50000

// ---------------------------------------------------------------------------
// Edge pass: build per-graph 10x10 adjacency counts (exact integer floats ->
// atomic order-independent -> deterministic).
__global__ __launch_bounds__(256) void k_edges(const int* __restrict__ ei,
                                               float* __restrict__ adj) {
    int e = blockIdx.x * 256 + threadIdx.x;
    if (e >= NEDGES) return;
    int s = ei[e];
    int d = ei[NEDGES + e];
    int g  = d / 10;
    int ls = s - (s / 10) * 10;
    int ld = d - g * 10;
    atomicAdd(&adj[(size_t)g * 100 + ld * 10 + ls], 1.0f);
}

// deg[dst] = row-sum of adj + 1 (self loop); dinv = rsqrt(deg)
__global__ __launch_bounds__(256) void k_dinv(const float* __restrict__ adj,
                                              float* __restrict__ dinv) {
    int i = blockIdx.x * 256 + threadIdx.x;
    if (i >= NNODES) return;
    int g = i / 10, ld = i - g * 10;
    const float* row = adj + (size_t)g * 100 + ld * 10;
    float deg = 1.0f;
#pragma unroll
    for (int ls = 0; ls < 10; ++ls) deg += row[ls];
    dinv[i] = rsqrtf(deg);
}

// ---------------------------------------------------------------------------
// Swizzle a [64,64] weight matrix (rows >= rlimit treated as zero) into the
// exact per-lane B-fragment layout for V_WMMA_F32_16X16X4_F32:
//   F2[(kk*4+nt)*32 + lane] = { W[kk*4+hi*2  ][nt*16+n],
//                               W[kk*4+hi*2+1][nt*16+n] }   lane = hi*16+n
// so the GEMM inner loop reads one conflict-free ds_load_b64 per fragment.
__global__ __launch_bounds__(256) void k_swizzleW(const float* __restrict__ W,
                                                  float* __restrict__ F,
                                                  int rlimit) {
    int tid = blockIdx.x * 256 + threadIdx.x;
    if (tid >= 2048) return;
    int lane = tid & 31;
    int frag = tid >> 5;          // kk*4 + nt
    int kk = frag >> 2;
    int nt = frag & 3;
    int hi = lane >> 4;
    int n  = lane & 15;
    int k0 = kk * 4 + hi * 2;
    int col = nt * 16 + n;
    float x0 = (k0     < rlimit) ? W[k0 * 64 + col]       : 0.0f;
    float x1 = (k0 + 1 < rlimit) ? W[(k0 + 1) * 64 + col] : 0.0f;
    F[2 * tid]     = x0;
    F[2 * tid + 1] = x1;
}

// Embedding gather -> h0 [N,64] (cols 41..63 zero). float4-quad per thread.
__global__ __launch_bounds__(256) void k_embed(const int* __restrict__ x,
                                               const float* __restrict__ ct,
                                               const float* __restrict__ rt,
                                               float* __restrict__ H) {
    long tid = (long)blockIdx.x * 256 + threadIdx.x;
    long node = tid >> 4;
    if (node >= NNODES) return;
    int q = (int)(tid & 15);
    int c0 = q * 4;
    float4 v; v.x = 0.f; v.y = 0.f; v.z = 0.f; v.w = 0.f;
    if (c0 < 32) {
        int champ = x[node * 3];
        v = *(const float4*)(ct + (size_t)champ * 32 + c0);
    } else if (c0 < 40) {
        int role = x[node * 3 + 1];
        v = *(const float4*)(rt + (size_t)role * 8 + (c0 - 32));
    } else if (c0 == 40) {
        v.x = (float)x[node * 3 + 2];
    }
    *(float4*)(H + node * 64 + c0) = v;
}

// ---------------------------------------------------------------------------
// Y[N,64] = X[N,64] @ W[64,64] via V_WMMA_F32_16X16X4_F32.
// W arrives pre-swizzled into per-lane fragments (see k_swizzleW).
// 8 waves/block, one 16-row tile per wave.
__global__ __launch_bounds__(256) void k_gemm64(const float* __restrict__ X,
                                                const float* __restrict__ Wf,
                                                float* __restrict__ Y,
                                                int nrows) {
    __shared__ float Wl[4096];
    int tid = threadIdx.x;
    {   // cooperative copy of swizzled fragments (16 KB)
        const float4* src = (const float4*)Wf;
        float4* dst = (float4*)Wl;
        for (int i = tid; i < 1024; i += 256) dst[i] = src[i];
    }
    __syncthreads();

    int wave = tid >> 5;
    int lane = tid & 31;
    int m  = lane & 15;
    int hi = lane >> 4;            // half-wave selects K sub-pair
    int base = blockIdx.x * 128 + wave * 16;
    if (base >= nrows) return;     // wave-uniform: EXEC stays all-ones for WMMA

    const float* arow = X + (size_t)(base + m) * 64;
    const v2f* Wfrag = (const v2f*)Wl;   // indexed [(kk*4+nt)*32 + lane]
    v8f acc[4];
    acc[0] = (v8f){}; acc[1] = (v8f){}; acc[2] = (v8f){}; acc[3] = (v8f){};

#pragma unroll
    for (int kk = 0; kk < 16; ++kk) {
        int k0 = kk * 4 + hi * 2;
        v2f a;
        a.x = arow[k0];
        a.y = arow[k0 + 1];
#pragma unroll
        for (int nt = 0; nt < 4; ++nt) {
            v2f b = Wfrag[(kk * 4 + nt) * 32 + lane];
            acc[nt] = __builtin_amdgcn_wmma_f32_16x16x4_f32(
                false, a, false, b, (short)0, acc[nt], false, false);
        }
    }

    // C/D layout: vgpr r holds row (r + hi*8), col = lane&15 (+ nt*16)
    float* yb = Y + (size_t)(base + hi * 8) * 64 + m;
#pragma unroll
    for (int r = 0; r < 8; ++r) {
        float* yr = yb + (size_t)r * 64;
        yr[0]  = acc[0][r];
        yr[16] = acc[1][r];
        yr[32] = acc[2][r];
        yr[48] = acc[3][r];
    }
}

// ---------------------------------------------------------------------------
// Per-graph normalized aggregation, in place:
//   out[ld] = bias + sum_ls (cnt[ld][ls]*dinv_ls*dinv_ld + (ls==ld)*dinv_ld^2) * hw[ls]
// 4 graphs per 256-thread block; 64 threads (one per feature col) per graph.
__global__ __launch_bounds__(256) void k_agg(float* __restrict__ HW,
                                             const float* __restrict__ adj,
                                             const float* __restrict__ dinv,
                                             const float* __restrict__ bias,
                                             int do_relu) {
    __shared__ float hws[4][10][64];
    __shared__ float co[4][100];
    __shared__ float dl[4][10];
    int tid = threadIdx.x;
    int sub = tid >> 6;
    int t   = tid & 63;
    int gi  = blockIdx.x * 4 + sub;
    bool ok = (gi < NGRAPH);
    if (ok) {
        const float* hwg = HW + (size_t)gi * 640;
#pragma unroll
        for (int ls = 0; ls < 10; ++ls) hws[sub][ls][t] = hwg[ls * 64 + t];
        if (t < 10) dl[sub][t] = dinv[gi * 10 + t];
    }
    __syncthreads();
    if (ok && t < 50) {
#pragma unroll
        for (int j = 0; j < 2; ++j) {
            int idx = t + 50 * j;
            int ld = idx / 10, ls = idx - ld * 10;
            float c = adj[(size_t)gi * 100 + idx] * dl[sub][ls] * dl[sub][ld];
            if (ls == ld) c += dl[sub][ld] * dl[sub][ld];
            co[sub][idx] = c;
        }
    }
    __syncthreads();
    if (ok) {
        float* hwg = HW + (size_t)gi * 640;
        float bv = bias[t];
#pragma unroll
        for (int ld = 0; ld < 10; ++ld) {
            float acc = bv;
#pragma unroll
            for (int ls = 0; ls < 10; ++ls)
                acc += co[sub][ld * 10 + ls] * hws[sub][ls][t];
            if (do_relu) acc = fmaxf(acc, 0.0f);
            hwg[ld * 64 + t] = acc;
        }
    }
}

// ---------------------------------------------------------------------------
// Column sum / sumsq partials (deterministic two-level reduction).
__global__ __launch_bounds__(256) void k_colstats(const float* __restrict__ X,
                                                  int nrows, int C, int rowsPerBlock,
                                                  float* __restrict__ partS,
                                                  float* __restrict__ partQ) {
    __shared__ float rs[256], rq[256];
    int tid = threadIdx.x;
    int tpc = 256 / C;
    int c = tid % C;
    int slice = tid / C;
    long r0 = (long)blockIdx.x * rowsPerBlock + slice;
    long rend = (long)(blockIdx.x + 1) * rowsPerBlock;
    if (rend > nrows) rend = nrows;
    float s = 0.f, q = 0.f;
    for (long r = r0; r < rend; r += tpc) {
        float v = X[r * C + c];
        s += v; q += v * v;
    }
    rs[tid] = s; rq[tid] = q;
    __syncthreads();
    if (tid < C) {
        float S = 0.f, Q = 0.f;
        for (int j = 0; j < tpc; ++j) { S += rs[c + j * C]; Q += rq[c + j * C]; }
        partS[(size_t)blockIdx.x * C + c] = S;
        partQ[(size_t)blockIdx.x * C + c] = Q;
    }
}

// Combine partials -> BN scale/shift per column.
__global__ void k_bnfin(const float* __restrict__ partS, const float* __restrict__ partQ,
                        int nblocks, int C, float invN,
                        const float* __restrict__ gamma, const float* __restrict__ beta,
                        float* __restrict__ scale, float* __restrict__ shift) {
    int c = threadIdx.x;
    if (c >= C) return;
    float S = 0.f, Q = 0.f;
    for (int b = 0; b < nblocks; ++b) { S += partS[(size_t)b * C + c]; Q += partQ[(size_t)b * C + c]; }
    float mu  = S * invN;
    float var = Q * invN - mu * mu;
    float sc  = gamma[c] * rsqrtf(var + 1e-5f);
    scale[c] = sc;
    shift[c] = beta[c] - mu * sc;
}

// h1 = relu(x*scale + shift), float4 quads over [N,64]
__global__ __launch_bounds__(256) void k_bnapply(const float4* __restrict__ X,
                                                 const float* __restrict__ scale,
                                                 const float* __restrict__ shift,
                                                 float4* __restrict__ Y, long nquads) {
    long i = (long)blockIdx.x * 256 + threadIdx.x;
    if (i >= nquads) return;
    int c = (int)((i & 15) << 2);
    float4 v = X[i];
    float4 sc = *(const float4*)(scale + c);
    float4 sh = *(const float4*)(shift + c);
    float4 o;
    o.x = fmaxf(v.x * sc.x + sh.x, 0.f);
    o.y = fmaxf(v.y * sc.y + sh.y, 0.f);
    o.z = fmaxf(v.z * sc.z + sh.z, 0.f);
    o.w = fmaxf(v.w * sc.w + sh.w, 0.f);
    Y[i] = o;
}

// Global mean pool (10 nodes/graph, count fixed).
__global__ __launch_bounds__(256) void k_pool(const float* __restrict__ H,
                                              float* __restrict__ g) {
    long tid = (long)blockIdx.x * 256 + threadIdx.x;
    int gi = (int)(tid >> 6);
    if (gi >= NGRAPH) return;
    int c = (int)(tid & 63);
    const float* base = H + (size_t)gi * 640 + c;
    float s = 0.f;
#pragma unroll
    for (int ls = 0; ls < 10; ++ls) s += base[ls * 64];
    g[(size_t)gi * 64 + c] = s * 0.1f;
}

// z = g @ fc1_W + fc1_b   [G,64] x [64,32]
__global__ __launch_bounds__(256) void k_fc1(const float* __restrict__ g,
                                             const float* __restrict__ W,
                                             const float* __restrict__ b,
                                             float* __restrict__ z) {
    long tid = (long)blockIdx.x * 256 + threadIdx.x;
    int gi = (int)(tid >> 5);
    if (gi >= NGRAPH) return;
    int c = (int)(tid & 31);
    const float* row = g + (size_t)gi * 64;
    float acc = b[c];
#pragma unroll
    for (int k = 0; k < 64; ++k) acc += row[k] * W[k * 32 + c];
    z[(size_t)gi * 32 + c] = acc;
}

// out = sigmoid(relu(bn2(z)) @ fc2_W + fc2_b)
__global__ __launch_bounds__(256) void k_final(const float* __restrict__ z,
                                               const float* __restrict__ scale2,
                                               const float* __restrict__ shift2,
                                               const float* __restrict__ w2,
                                               const float* __restrict__ b2,
                                               float* __restrict__ out) {
    int gi = blockIdx.x * 256 + threadIdx.x;
    if (gi >= NGRAPH) return;
    const float* row = z + (size_t)gi * 32;
    float acc = b2[0];
#pragma unroll
    for (int k = 0; k < 32; ++k) {
        float h = fmaxf(row[k] * scale2[k] + shift2[k], 0.f);
        acc += h * w2[k];
    }
    out[gi] = 1.0f / (1.0f + expf(-acc));
}

// ---------------------------------------------------------------------------
extern "C" void kernel_launch(void* const* d_in, const int* in_sizes, int n_in,
                              void* d_out, int out_size, void* d_ws, size_t ws_size,
                              hipStream_t stream) {
    const int*   x     = (const int*)d_in[0];
    const int*   ei    = (const int*)d_in[1];
    const float* champ = (const float*)d_in[3];
    const float* role  = (const float*)d_in[4];
    const float* W1    = (const float*)d_in[5];
    const float* b1    = (const float*)d_in[6];
    const float* g1    = (const float*)d_in[7];
    const float* be1   = (const float*)d_in[8];
    const float* W2    = (const float*)d_in[9];
    const float* b2    = (const float*)d_in[10];
    const float* f1W   = (const float*)d_in[11];
    const float* f1b   = (const float*)d_in[12];
    const float* g2    = (const float*)d_in[13];
    const float* be2   = (const float*)d_in[14];
    const float* f2W   = (const float*)d_in[15];
    const float* f2b   = (const float*)d_in[16];
    float* out = (float*)d_out;

    float* ws    = (float*)d_ws;
    float* adj   = ws;                       // 5,000,000
    float* dinv  = adj   + 5000000;          // 500,000
    float* Wf1   = dinv  + 500000;           // 4,096  (swizzled, zero-padded W1)
    float* Wf2   = Wf1   + 4096;             // 4,096  (swizzled W2)
    float* ssh   = Wf2   + 4096;             // 256: scale1/shift1 @0/64, scale2/shift2 @128/160
    float* partS = ssh   + 256;              // 16,384
    float* partQ = partS + 16384;            // 16,384
    float* gpool = partQ + 16384;            // 3,200,000
    float* z     = gpool + 3200000;          // 1,600,000
    float* A     = z     + 1600000;          // 32,000,000  (h0 / h1)
    float* B     = A     + 32000000;         // 32,000,000  (hw / agg)

    hipMemsetAsync(adj, 0, 5000000 * sizeof(float), stream);

    const int sb1 = (NNODES + 2047) / 2048;   // BN1 partial blocks
    const int sb2 = (NGRAPH + 2047) / 2048;   // BN2 partial blocks

    k_edges   <<<(NEDGES + 255) / 256, 256, 0, stream>>>(ei, adj);
    k_dinv    <<<(NNODES + 255) / 256, 256, 0, stream>>>(adj, dinv);
    k_swizzleW<<<8, 256, 0, stream>>>(W1, Wf1, 41);
    k_swizzleW<<<8, 256, 0, stream>>>(W2, Wf2, 64);
    k_embed   <<<(NNODES * 16 + 255) / 256, 256, 0, stream>>>(x, champ, role, A);

    // Layer 1: GEMM (WMMA) -> graph aggregation (+b1) -> BN -> ReLU
    k_gemm64<<<(NNODES / 16 + 7) / 8, 256, 0, stream>>>(A, Wf1, B, NNODES);
    k_agg   <<<(NGRAPH + 3) / 4, 256, 0, stream>>>(B, adj, dinv, b1, 0);
    k_colstats<<<sb1, 256, 0, stream>>>(B, NNODES, 64, 2048, partS, partQ);
    k_bnfin <<<1, 64, 0, stream>>>(partS, partQ, sb1, 64, 1.0f / NNODES, g1, be1, ssh, ssh + 64);
    k_bnapply<<<(NNODES * 16 + 255) / 256, 256, 0, stream>>>((const float4*)B, ssh, ssh + 64,
                                                             (float4*)A, (long)NNODES * 16);

    // Layer 2: GEMM (WMMA) -> graph aggregation (+b2, ReLU)
    k_gemm64<<<(NNODES / 16 + 7) / 8, 256, 0, stream>>>(A, Wf2, B, NNODES);
    k_agg   <<<(NGRAPH + 3) / 4, 256, 0, stream>>>(B, adj, dinv, b2, 1);

    // Head: pool -> fc1 -> BN2 -> ReLU -> fc2 -> sigmoid
    k_pool  <<<(NGRAPH * 64 + 255) / 256, 256, 0, stream>>>(B, gpool);
    k_fc1   <<<(NGRAPH * 32 + 255) / 256, 256, 0, stream>>>(gpool, f1W, f1b, z);
    k_colstats<<<sb2, 256, 0, stream>>>(z, NGRAPH, 32, 2048, partS, partQ);
    k_bnfin <<<1, 32, 0, stream>>>(partS, partQ, sb2, 32, 1.0f / NGRAPH, g2, be2, ssh + 128, ssh + 160);
    k_final <<<(NGRAPH + 255) / 256, 256, 0, stream>>>(z, ssh + 128, ssh + 160, f2W, f2b, out);
}